// SAGE_29841432773054
// MI455X (gfx1250) — compile-verified
//
#include <hip/hip_runtime.h>

#define N_NODES   100000
#define N_EDGES   1600000
#define IN_FEATS  128
#define N_HIDDEN  128
#define N_CLASSES 47

typedef __attribute__((ext_vector_type(2))) float v2f;
typedef __attribute__((ext_vector_type(8))) float v8f;

// ---------------------------------------------------------------------------
// Degree computation: deg[dst[e]] += 1
// ---------------------------------------------------------------------------
__global__ void sage_deg_kernel(const int* __restrict__ dst,
                                float* __restrict__ deg) {
    int i = blockIdx.x * blockDim.x + threadIdx.x;
    if (i < N_EDGES) {
        unsafeAtomicAdd(&deg[dst[i]], 1.0f);
    }
}

// deg -> 1 / max(deg, 1)
__global__ void sage_invdeg_kernel(float* __restrict__ deg) {
    int i = blockIdx.x * blockDim.x + threadIdx.x;
    if (i < N_NODES) {
        deg[i] = 1.0f / fmaxf(deg[i], 1.0f);
    }
}

// ---------------------------------------------------------------------------
// Edge scatter: accum[dst, :] += h[src, :]  (one wave per edge, float4/lane)
// ---------------------------------------------------------------------------
__global__ void sage_scatter_kernel(const float* __restrict__ h,
                                    const int* __restrict__ src,
                                    const int* __restrict__ dst,
                                    float* __restrict__ accum) {
    int gid  = blockIdx.x * blockDim.x + threadIdx.x;
    int edge = gid >> 5;
    int lane = threadIdx.x & 31;
    if (edge >= N_EDGES) return;

    int s = src[edge];
    int d = dst[edge];
    const float4* hv = reinterpret_cast<const float4*>(h + (size_t)s * 128);
    float4 v = hv[lane];                       // coalesced 512B per wave
    float* out = accum + (size_t)d * 128 + lane * 4;
    unsafeAtomicAdd(out + 0, v.x);
    unsafeAtomicAdd(out + 1, v.y);
    unsafeAtomicAdd(out + 2, v.z);
    unsafeAtomicAdd(out + 3, v.w);
}

// ---------------------------------------------------------------------------
// Weight pack: frag[(kg*2+hh)*doutp + n] (float2) = { W[kg*4+hh*2][n],
//                                                     W[kg*4+hh*2+1][n] }
// Zero-pads columns n >= dout so the GEMM hot loop needs no masking.
// ---------------------------------------------------------------------------
__global__ void sage_pack_kernel(const float* __restrict__ W,
                                 float* __restrict__ frag,
                                 int dout, int doutp) {
    int i = blockIdx.x * blockDim.x + threadIdx.x;  // over 64*doutp float2's
    int total = 64 * doutp;
    if (i >= total) return;
    int n = i % doutp;
    int g = i / doutp;               // kg*2 + hh, 0..63
    int k = (g >> 1) * 4 + (g & 1) * 2;
    float a = 0.0f, b = 0.0f;
    if (n < dout) {
        a = W[(size_t)k * dout + n];
        b = W[(size_t)(k + 1) * dout + n];
    }
    frag[2 * i]     = a;
    frag[2 * i + 1] = b;
}

// ---------------------------------------------------------------------------
// Fused dual GEMM + mean-normalize + optional ReLU, fp32 WMMA 16x16x4.
// One block = one 16-row tile; wave w handles column tiles w, w+8, ...
//   out[16,DOUT] = h[16,128] @ Wself + (accum[16,128]*invdeg) @ Wneigh
// A tiles staged in LDS (conflict-free padded rows); invdeg folded into the
// staged neighbor tile; B comes from pre-packed fragment buffers with
// compile-time immediate offsets.
// ---------------------------------------------------------------------------
template <int DOUTP, int DOUT, int TILES_N, int RELU>
__global__ __launch_bounds__(256) void sage_gemm_kernel(
    const float* __restrict__ h,
    const float* __restrict__ accum,
    const float* __restrict__ invdeg,
    const float* __restrict__ fragS,
    const float* __restrict__ fragN,
    float* __restrict__ out) {
    constexpr int LROW = 132;              // padded row stride: 528B, 8B-aligned
    __shared__ float lh[16 * LROW];
    __shared__ float lg[16 * LROW];

    const int t    = threadIdx.x;
    const int row0 = blockIdx.x * 16;

    // ---- stage 16x128 tiles of h and (accum * invdeg) into LDS ----
#pragma unroll
    for (int i = t; i < 512; i += 256) {
        int row = i >> 5;                  // 0..15
        int c4  = (i & 31) * 4;            // 0,4,...,124
        size_t goff = (size_t)(row0 + row) * 128 + c4;
        float4 hv = *(const float4*)(h + goff);
        float4 gv = *(const float4*)(accum + goff);
        float idg = invdeg[row0 + row];
        gv.x *= idg; gv.y *= idg; gv.z *= idg; gv.w *= idg;
        *(float4*)(&lh[row * LROW + c4]) = hv;
        *(float4*)(&lg[row * LROW + c4]) = gv;
    }
    __syncthreads();

    const int wid  = t >> 5;
    const int lane = t & 31;
    const int hh   = lane >> 4;            // 0/1: which K-half of the group
    const int l15  = lane & 15;

    // A fragment pointers (float2 units): lane reads A[M=l15][K=kg*4+hh*2+{0,1}]
    const v2f* aH = (const v2f*)lh + (l15 * (LROW / 2) + hh);
    const v2f* aG = (const v2f*)lg + (l15 * (LROW / 2) + hh);

    for (int tn = wid; tn < TILES_N; tn += 8) {
        int ncol = tn * 16 + l15;
        const v2f* bS = (const v2f*)fragS + (hh * DOUTP + ncol);
        const v2f* bN = (const v2f*)fragN + (hh * DOUTP + ncol);

        v8f c = {};
#pragma unroll 8
        for (int kg = 0; kg < 32; ++kg) {
            c = __builtin_amdgcn_wmma_f32_16x16x4_f32(
                false, aH[kg * 2], false, bS[kg * 2 * DOUTP],
                (short)0, c, false, false);
        }
#pragma unroll 8
        for (int kg = 0; kg < 32; ++kg) {
            c = __builtin_amdgcn_wmma_f32_16x16x4_f32(
                false, aG[kg * 2], false, bN[kg * 2 * DOUTP],
                (short)0, c, false, false);
        }

        // C/D layout: VGPR r -> M=r (lanes 0-15), M=r+8 (lanes 16-31); N=l15
        int srow0 = row0 + hh * 8;
        if (DOUT == DOUTP || ncol < DOUT) {
#pragma unroll
            for (int r = 0; r < 8; ++r) {
                float v = c[r];
                if (RELU) v = fmaxf(v, 0.0f);
                out[(size_t)(srow0 + r) * DOUT + ncol] = v;
            }
        }
    }
}

// ---------------------------------------------------------------------------
// Host launcher
// ---------------------------------------------------------------------------
extern "C" void kernel_launch(void* const* d_in, const int* in_sizes, int n_in,
                              void* d_out, int out_size, void* d_ws, size_t ws_size,
                              hipStream_t stream) {
    const float* x   = (const float*)d_in[0];
    const float* Ws0 = (const float*)d_in[1];
    const float* Wn0 = (const float*)d_in[2];
    const float* Ws1 = (const float*)d_in[3];
    const float* Wn1 = (const float*)d_in[4];
    const float* Ws2 = (const float*)d_in[5];
    const float* Wn2 = (const float*)d_in[6];
    const int*   src = (const int*)d_in[7];
    const int*   dst = (const int*)d_in[8];

    const size_t NH     = (size_t)N_NODES * 128;
    const size_t FRAG_H = 64 * 128 * 2;    // floats, hidden-layer frag (64KB)
    const size_t FRAG_C = 64 * 48 * 2;     // floats, class-layer frag  (24KB)

    float* accum  = (float*)d_ws;          // [N,128] neighbor-sum accumulator
    float* hA     = accum + NH;            // [N,128] hidden ping
    float* hB     = hA + NH;               // [N,128] hidden pong
    float* invdeg = hB + NH;               // [N]
    float* fS0 = invdeg + N_NODES;
    float* fN0 = fS0 + FRAG_H;
    float* fS1 = fN0 + FRAG_H;
    float* fN1 = fS1 + FRAG_H;
    float* fS2 = fN1 + FRAG_H;
    float* fN2 = fS2 + FRAG_C;

    const int TPB = 256;
    const unsigned edge_blocks    = (N_EDGES + TPB - 1) / TPB;
    const unsigned node_blocks    = (N_NODES + TPB - 1) / TPB;
    const unsigned scatter_blocks = (unsigned)(((size_t)N_EDGES * 32 + TPB - 1) / TPB);
    const unsigned packH_blocks   = (64 * 128 + TPB - 1) / TPB;
    const unsigned packC_blocks   = (64 * 48 + TPB - 1) / TPB;
    const unsigned gemm_blocks    = N_NODES / 16;   // 6250 (exact)

    // Degrees (same for all layers)
    hipMemsetAsync(invdeg, 0, N_NODES * sizeof(float), stream);
    sage_deg_kernel<<<edge_blocks, TPB, 0, stream>>>(dst, invdeg);
    sage_invdeg_kernel<<<node_blocks, TPB, 0, stream>>>(invdeg);

    // Pack all weight matrices into B-fragment layout
    sage_pack_kernel<<<packH_blocks, TPB, 0, stream>>>(Ws0, fS0, 128, 128);
    sage_pack_kernel<<<packH_blocks, TPB, 0, stream>>>(Wn0, fN0, 128, 128);
    sage_pack_kernel<<<packH_blocks, TPB, 0, stream>>>(Ws1, fS1, 128, 128);
    sage_pack_kernel<<<packH_blocks, TPB, 0, stream>>>(Wn1, fN1, 128, 128);
    sage_pack_kernel<<<packC_blocks, TPB, 0, stream>>>(Ws2, fS2, 47, 48);
    sage_pack_kernel<<<packC_blocks, TPB, 0, stream>>>(Wn2, fN2, 47, 48);

    // ---- Layer 0: x -> hA (ReLU) ----
    hipMemsetAsync(accum, 0, NH * sizeof(float), stream);
    sage_scatter_kernel<<<scatter_blocks, TPB, 0, stream>>>(x, src, dst, accum);
    sage_gemm_kernel<128, 128, 8, 1><<<gemm_blocks, TPB, 0, stream>>>(
        x, accum, invdeg, fS0, fN0, hA);

    // ---- Layer 1: hA -> hB (ReLU) ----
    hipMemsetAsync(accum, 0, NH * sizeof(float), stream);
    sage_scatter_kernel<<<scatter_blocks, TPB, 0, stream>>>(hA, src, dst, accum);
    sage_gemm_kernel<128, 128, 8, 1><<<gemm_blocks, TPB, 0, stream>>>(
        hA, accum, invdeg, fS1, fN1, hB);

    // ---- Layer 2: hB -> d_out (no ReLU) ----
    hipMemsetAsync(accum, 0, NH * sizeof(float), stream);
    sage_scatter_kernel<<<scatter_blocks, TPB, 0, stream>>>(hB, src, dst, accum);
    sage_gemm_kernel<48, 47, 3, 0><<<gemm_blocks, TPB, 0, stream>>>(
        hB, accum, invdeg, fS2, fN2, (float*)d_out);
}